// RMultiHeadGraphAttention2m_52716428591538
// MI455X (gfx1250) — compile-verified
//
#include <hip/hip_runtime.h>
#include <hip/hip_bf16.h>

typedef __attribute__((ext_vector_type(2))) float v2f;
typedef __attribute__((ext_vector_type(8))) float v8f;

#define F_OUT     64
#define NEG_SLOPE 0.2f
#define TILE      16   // edges per wave per WMMA batch

__device__ __forceinline__ float lrelu_neg_exp(float s) {
    // exp(-leaky_relu(s, 0.2))
    float lr = (s > 0.0f) ? s : NEG_SLOPE * s;
    return __expf(-lr);
}

// -----------------------------------------------------------------------------
// Fused edge kernel: one wave processes TILE=16 edges.
// Phase 1 (WMMA): scores[16 edges][2 heads] = A_src(16x64) * Bsrc(64x2)
//                                           + A_rel(16x64) * Brel(64x2)
//   chained as 32x V_WMMA_F32_16X16X4_F32 (K=4 per op).
//   Bsrc columns: [a00 | w .* a10], Brel columns: [a01 | a11], cols 2..15 = 0.
// Phase 2 (scatter): per edge, broadcast s0/s1 from the D-matrix layout,
//   all 32 lanes scatter (src - rel)*e0 and (src.*w - rel)*e1 plus rowsums
//   with global_atomic_add_f32.
// -----------------------------------------------------------------------------
__global__ void __launch_bounds__(256)
rgat_edge_kernel(const float* __restrict__ h,
                 const float* __restrict__ rtab,
                 const int*   __restrict__ Adst,
                 const int*   __restrict__ Arel,
                 const int*   __restrict__ Asrc,
                 const float* __restrict__ w,     // 64
                 const float* __restrict__ a,     // 256: a00,a01,a10,a11
                 float*       __restrict__ out,   // [2][N][64], pre-zeroed
                 float*       __restrict__ rowsum,// [2][N],    pre-zeroed
                 int E, int N)
{
    const int lane   = threadIdx.x & 31;
    const int wid    = (blockIdx.x * blockDim.x + threadIdx.x) >> 5;
    const int nwaves = (gridDim.x * blockDim.x) >> 5;
    const int ntiles = (E + TILE - 1) / TILE;

    const float* a00 = a;
    const float* a01 = a + 64;
    const float* a10 = a + 128;
    const float* a11 = a + 192;

    // ---- loop-invariant B fragments (registers) ----------------------------
    // B (4x16, KxN) fragment layout mirrors A: lanes 0-15 hold K={4t,4t+1},
    // lanes 16-31 hold K={4t+2,4t+3}; N = lane & 15.
    const int n     = lane & 15;
    const int khalf = (lane < 16) ? 0 : 2;
    v2f bsrc[TILE], brel[TILE];
#pragma unroll
    for (int t = 0; t < TILE; ++t) {
        const int k0 = 4 * t + khalf;
        float sx = 0.f, sy = 0.f, rx = 0.f, ry = 0.f;
        if (n == 0) {               // column 0 -> head 0
            sx = a00[k0];           sy = a00[k0 + 1];
            rx = a01[k0];           ry = a01[k0 + 1];
        } else if (n == 1) {        // column 1 -> head 1 (fold w into a10)
            sx = w[k0]     * a10[k0];
            sy = w[k0 + 1] * a10[k0 + 1];
            rx = a11[k0];           ry = a11[k0 + 1];
        }
        bsrc[t] = (v2f){sx, sy};
        brel[t] = (v2f){rx, ry};
    }
    // per-lane slice of w for the head-1 message (features 2*lane, 2*lane+1)
    const v2f wv = *(const v2f*)(w + 2 * lane);

    for (int tile = wid; tile < ntiles; tile += nwaves) {
        const long base = (long)tile * TILE;
        int e = (int)base + (lane & 15);
        e = (e < E) ? e : (E - 1);                 // clamp: keep EXEC all-ones
        const int srcrow = Asrc[e];
        const int relrow = Arel[e];
        const int dstrow = Adst[e];

        const float* sp = h    + (long)srcrow * F_OUT + khalf;
        const float* rp = rtab + (long)relrow * F_OUT + khalf;

        // ---- phase 1: 32 chained f32 WMMAs over K=64 -----------------------
        v8f acc = {};
#pragma unroll
        for (int t = 0; t < TILE; ++t) {
            const v2f av = *(const v2f*)(sp + 4 * t);
            const v2f bv = *(const v2f*)(rp + 4 * t);
            acc = __builtin_amdgcn_wmma_f32_16x16x4_f32(
                      false, av, false, bsrc[t], (short)0, acc, false, false);
            acc = __builtin_amdgcn_wmma_f32_16x16x4_f32(
                      false, bv, false, brel[t], (short)0, acc, false, false);
        }

        // ---- phase 2: per-edge scatter -------------------------------------
        // D layout: VGPR r holds M=r (lanes 0-15, N=lane) and M=r+8
        // (lanes 16-31, N=lane-16).  M=edge, N=0 -> s0, N=1 -> s1.
#pragma unroll
        for (int ei = 0; ei < TILE; ++ei) {
            const int srcLaneBase = (ei < 8) ? 0 : 16;
            const float s0 = __shfl(acc[ei & 7], srcLaneBase + 0, 32);
            const float s1 = __shfl(acc[ei & 7], srcLaneBase + 1, 32);
            const int  de  = __shfl(dstrow, ei, 32);
            const int  se  = __shfl(srcrow, ei, 32);
            const int  re  = __shfl(relrow, ei, 32);
            if (base + ei < E) {
                const float e0 = lrelu_neg_exp(s0);
                const float e1 = lrelu_neg_exp(s1);
                const v2f sv = *(const v2f*)(h    + (long)se * F_OUT + 2 * lane);
                const v2f rv = *(const v2f*)(rtab + (long)re * F_OUT + 2 * lane);
                float* o0 = out + (long)de * F_OUT + 2 * lane;
                float* o1 = o0 + (long)N * F_OUT;
                atomicAdd(o0,     (sv.x - rv.x) * e0);
                atomicAdd(o0 + 1, (sv.y - rv.y) * e0);
                atomicAdd(o1,     (sv.x * wv.x - rv.x) * e1);
                atomicAdd(o1 + 1, (sv.y * wv.y - rv.y) * e1);
                if (lane == 0) {
                    atomicAdd(rowsum + de,     e0);
                    atomicAdd(rowsum + N + de, e1);
                }
            }
        }
    }
}

// out[head][node][f] /= rowsum[head][node]
__global__ void __launch_bounds__(256)
rgat_norm_kernel(float* __restrict__ out, const float* __restrict__ rowsum,
                 int N, int total)
{
    int idx = blockIdx.x * blockDim.x + threadIdx.x;
    if (idx < total) {
        const int node_h = idx >> 6;            // head*N + node
        out[idx] = out[idx] / rowsum[node_h];
    }
}

extern "C" void kernel_launch(void* const* d_in, const int* in_sizes, int n_in,
                              void* d_out, int out_size, void* d_ws, size_t ws_size,
                              hipStream_t stream) {
    const float* h    = (const float*)d_in[0];
    const float* rtab = (const float*)d_in[1];
    const int*   A    = (const int*)d_in[2];
    const float* w    = (const float*)d_in[3];
    const float* a    = (const float*)d_in[4];

    const int N = in_sizes[0] / F_OUT;
    const int E = in_sizes[2] / 3;
    const int* Adst = A;
    const int* Arel = A + E;
    const int* Asrc = A + 2 * (size_t)E;

    float* out    = (float*)d_out;
    float* rowsum = (float*)d_ws;              // 2*N floats

    hipMemsetAsync(d_out, 0, (size_t)out_size * sizeof(float), stream);
    hipMemsetAsync(rowsum, 0, (size_t)2 * N * sizeof(float), stream);

    const int ntiles = (E + TILE - 1) / TILE;  // 100000
    const int wavesPerBlock = 8;               // 256 threads
    int blocks = (ntiles + wavesPerBlock - 1) / wavesPerBlock;
    rgat_edge_kernel<<<blocks, 256, 0, stream>>>(h, rtab, Adst, Arel, Asrc,
                                                 w, a, out, rowsum, E, N);

    const int total = out_size;                // 2*N*64
    rgat_norm_kernel<<<(total + 255) / 256, 256, 0, stream>>>(out, rowsum, N, total);
}